// ts_return_47596827574815
// MI455X (gfx1250) — compile-verified
//
#include <hip/hip_runtime.h>

// out[b,n,k] = X[b,n,10k+9] / X[b,n,10k] - 2.0f
// X: (64,512,4000) f32, out: (64,512,400) f32. Pure HBM-bandwidth kernel
// (~577 MB traffic => ~25us floor at 23.3 TB/s). Strategy: async-DMA each row
// into LDS with GLOBAL_LOAD_ASYNC_TO_LDS_B128 (contiguous 128b/lane), then do
// the stride-10-dword extraction from LDS (provably bank-conflict-free under
// 64-bank LDS), coalesced f32 store.

#define AS1 __attribute__((address_space(1)))
#define AS3 __attribute__((address_space(3)))

typedef int v4i __attribute__((ext_vector_type(4)));

#ifndef __has_builtin
#define __has_builtin(x) 0
#endif

__device__ __forceinline__ void async_copy_b128(const float* gsrc, float* ldst) {
#if __has_builtin(__builtin_amdgcn_global_load_async_to_lds_b128)
    // (global int4* src, lds int4* dst, imm offset, imm cpol)
    __builtin_amdgcn_global_load_async_to_lds_b128(
        (AS1 v4i*)(gsrc), (AS3 v4i*)(ldst), 0, 0);
#else
    // Fallback: ordinary 128-bit load + LDS store (still coalesced/correct).
    float4 v = *(const float4*)gsrc;
    *(float4*)ldst = v;
#endif
}

__device__ __forceinline__ void wait_async_all() {
#if __has_builtin(__builtin_amdgcn_s_wait_asynccnt)
    __builtin_amdgcn_s_wait_asynccnt(0);
#else
    asm volatile("s_wait_asynccnt 0x0" ::: "memory");
#endif
}

constexpr int T_LEN  = 4000;          // samples per row
constexpr int W_OUT  = 400;           // windows per row
constexpr int CHUNKS = T_LEN / 4;     // 1000 float4 chunks per row
constexpr int BLOCK  = 256;           // 8 wave32s

__global__ __launch_bounds__(BLOCK)
void ts_return_kernel(const float* __restrict__ X, float* __restrict__ out) {
    __shared__ float row[T_LEN];      // 16 KB of the 320 KB WGP LDS

    const int r = blockIdx.x;
    const float* src = X + (size_t)r * T_LEN;

    // Stage the whole row into LDS: 1000 x b128 async copies, 16B per lane,
    // fully contiguous across the wave (512B/wave per instruction).
    for (int c = threadIdx.x; c < CHUNKS; c += BLOCK) {
        async_copy_b128(src + c * 4, &row[c * 4]);
    }
    wait_async_all();     // each wave drains its own ASYNCcnt
    __syncthreads();      // ...then all waves' data is visible in LDS

    // Extract: dword indices 10k and 10k+9. 10k mod 64 is injective over 32
    // consecutive k (5 invertible mod 32) => no LDS bank conflicts.
    float* dst = out + (size_t)r * W_OUT;
    for (int k = threadIdx.x; k < W_OUT; k += BLOCK) {
        float first = row[10 * k];
        float last  = row[10 * k + 9];
        dst[k] = last / first - 2.0f;
    }
}

extern "C" void kernel_launch(void* const* d_in, const int* in_sizes, int n_in,
                              void* d_out, int out_size, void* d_ws, size_t ws_size,
                              hipStream_t stream) {
    const float* X  = (const float*)d_in[0];
    float* out      = (float*)d_out;
    const int rows  = in_sizes[0] / T_LEN;   // 64*512 = 32768
    hipLaunchKernelGGL(ts_return_kernel, dim3(rows), dim3(BLOCK), 0, stream,
                       X, out);
}